// ExpertParallelMoE_36318243454996
// MI455X (gfx1250) — compile-verified
//
#include <hip/hip_runtime.h>
#include <math.h>

#define DEV __device__ __forceinline__

typedef float v2f __attribute__((ext_vector_type(2)));
typedef float v8f __attribute__((ext_vector_type(8)));

constexpr int H = 768, E = 8, DFF = 3072, T = 2048;
constexpr int TILE_ROWS = 32;              // 2 M-tiles of 16 per block/wave
constexpr int RMAX = 4352;                 // 2*T + E*31 = 4344 -> padded to 32
constexpr int TILES_MAX = RMAX / TILE_ROWS; // 136

// ---- workspace layout (bytes) ----
constexpr size_t OFF_IDX2  = 0;                                  // T*2 ints
constexpr size_t OFF_ROWOF = OFF_IDX2  + (size_t)T * 2 * 4;      // T*2 ints
constexpr size_t OFF_R2T   = OFF_ROWOF + (size_t)T * 2 * 4;      // RMAX ints
constexpr size_t OFF_T2E   = OFF_R2T   + (size_t)RMAX * 4;       // 272 ints
constexpr size_t OFF_META  = OFF_T2E   + 272 * 4;                // 16 ints
constexpr size_t OFF_WTS   = OFF_META  + 16 * 4;                 // T*2 floats
constexpr size_t OFF_XG    = OFF_WTS   + (size_t)T * 2 * 4;      // RMAX*H floats
constexpr size_t OFF_MID   = OFF_XG    + (size_t)RMAX * H * 4;   // RMAX*DFF floats
constexpr size_t OFF_BUF2  = OFF_MID   + (size_t)RMAX * DFF * 4; // RMAX*H floats

DEV v8f wmma_f32(v2f a, v2f b, v8f c) {
  // D = A(16x4 f32) * B(4x16 f32) + C(16x16 f32)
  return __builtin_amdgcn_wmma_f32_16x16x4_f32(
      /*neg_a=*/false, a, /*neg_b=*/false, b,
      /*c_mod=*/(short)0, c, /*reuse_a=*/false, /*reuse_b=*/false);
}

DEV float gelu_exact(float v) {
  return 0.5f * v * (1.0f + erff(v * 0.7071067811865476f));
}

// ---------------- 1) gating: logits -> softmax -> top-2 -> renorm ----------------
__global__ void moe_gate_kernel(const float* __restrict__ x,
                                const float* __restrict__ Wg,
                                const float* __restrict__ bg,
                                int* __restrict__ idx2,
                                float* __restrict__ wts) {
  int t = blockIdx.x * blockDim.x + threadIdx.x;
  if (t >= T) return;
  float acc[E];
#pragma unroll
  for (int e = 0; e < E; ++e) acc[e] = bg[e];
  const float* xr = x + (size_t)t * H;
  for (int h = 0; h < H; ++h) {
    float xv = xr[h];
    const float* wr = Wg + (size_t)h * E;
#pragma unroll
    for (int e = 0; e < E; ++e) acc[e] += xv * wr[e];
  }
  float mx = acc[0];
#pragma unroll
  for (int e = 1; e < E; ++e) mx = fmaxf(mx, acc[e]);
  float p[E], s = 0.f;
#pragma unroll
  for (int e = 0; e < E; ++e) { p[e] = expf(acc[e] - mx); s += p[e]; }
  float inv = 1.f / s;
#pragma unroll
  for (int e = 0; e < E; ++e) p[e] *= inv;
  // top-2, first occurrence wins ties (matches lax.top_k)
  int i0 = 0; float b0 = p[0];
#pragma unroll
  for (int e = 1; e < E; ++e) if (p[e] > b0) { b0 = p[e]; i0 = e; }
  int i1 = -1; float b1 = -1.f;
#pragma unroll
  for (int e = 0; e < E; ++e) if (e != i0 && p[e] > b1) { b1 = p[e]; i1 = e; }
  float sw = 1.f / (b0 + b1);
  idx2[2 * t] = i0; idx2[2 * t + 1] = i1;
  wts[2 * t] = b0 * sw; wts[2 * t + 1] = b1 * sw;
}

// ---------------- 2) routing: deterministic per-expert row lists ----------------
__global__ void moe_route_kernel(const int* __restrict__ idx2,
                                 int* __restrict__ r2t,
                                 int* __restrict__ rowof,
                                 int* __restrict__ t2e,
                                 int* __restrict__ meta) {
  __shared__ int counts[E];
  __shared__ int starts[E + 1];
  int tid = threadIdx.x;  // 32 threads
  if (tid < E) {
    int c = 0;
    for (int t = 0; t < T; ++t) {
      if (idx2[2 * t] == tid || idx2[2 * t + 1] == tid) ++c;
    }
    counts[tid] = c;
  }
  __syncthreads();
  if (tid == 0) {
    int s = 0;
    for (int e = 0; e < E; ++e) { starts[e] = s; s += (counts[e] + 31) & ~31; }
    starts[E] = s;
    meta[0] = s;                // padded row count (multiple of 32)
    meta[1] = s / TILE_ROWS;    // active 32-row tiles
  }
  __syncthreads();
  for (int r = tid; r < RMAX; r += 32) r2t[r] = -1;
  for (int tt = tid; tt < TILES_MAX; tt += 32) t2e[tt] = -1;
  __syncthreads();
  if (tid < E) {
    int r = starts[tid];
    for (int t = 0; t < T; ++t) {
      int a = idx2[2 * t], b = idx2[2 * t + 1];
      if (a == tid)      { r2t[r] = t; rowof[2 * t] = r;     ++r; }
      else if (b == tid) { r2t[r] = t; rowof[2 * t + 1] = r; ++r; }
    }
    for (int tt = starts[tid] / TILE_ROWS; tt < starts[tid + 1] / TILE_ROWS; ++tt)
      t2e[tt] = tid;
  }
}

// ---------------- 3) gather X rows into expert-ordered Xg ----------------
__global__ void moe_gather_kernel(const float* __restrict__ x,
                                  const int* __restrict__ r2t,
                                  float* __restrict__ Xg) {
  int gid = blockIdx.x * blockDim.x + threadIdx.x;  // RMAX*H threads
  int r = gid / H, h = gid - r * H;
  int t = r2t[r];
  Xg[gid] = (t >= 0) ? x[(size_t)t * H + h] : 0.0f;
}

// ---------------- 4) GEMM1 (Xg @ W1[e]) + bias + exact GELU, f32 WMMA ----------------
// block = 256 threads = 8 waves; each wave computes a 32x32 output tile (2M x 2N).
__global__ void moe_gemm1_kernel(const float* __restrict__ Xg,
                                 const float* __restrict__ W1,
                                 const float* __restrict__ b1,
                                 float* __restrict__ Mid,
                                 const int* __restrict__ t2e,
                                 const int* __restrict__ meta) {
  int tile = blockIdx.x;
  if (tile >= meta[1]) return;         // uniform exit: EXEC stays all-1s for WMMA
  int e = t2e[tile];
  if (e < 0) return;

  int lane = threadIdx.x & 31;
  int wave = threadIdx.x >> 5;
  int m = lane & 15, half = lane >> 4;
  int rowBase = tile * TILE_ROWS;
  int col0 = blockIdx.y * 256 + wave * 32 + m;  // N-tiles at col0 and col0+16

  const float* A0 = Xg + (size_t)(rowBase + m) * H + 2 * half;
  const float* A1 = A0 + (size_t)16 * H;
  const float* B = W1 + (size_t)e * H * DFF + (size_t)(2 * half) * DFF + col0;

  v8f acc00 = {0, 0, 0, 0, 0, 0, 0, 0};
  v8f acc01 = {0, 0, 0, 0, 0, 0, 0, 0};
  v8f acc10 = {0, 0, 0, 0, 0, 0, 0, 0};
  v8f acc11 = {0, 0, 0, 0, 0, 0, 0, 0};
#pragma unroll 4
  for (int k0 = 0; k0 < H; k0 += 4) {
    v2f a0 = *(const v2f*)(A0 + k0);         // A[m][k0+2*half + {0,1}]
    v2f a1 = *(const v2f*)(A1 + k0);         // A[m+16][...]
    const float* Bp = B + (size_t)k0 * DFF;  // B[k0+2*half + j][col]
    v2f bv0; bv0.x = Bp[0];  bv0.y = Bp[DFF];
    v2f bv1; bv1.x = Bp[16]; bv1.y = Bp[DFF + 16];
    __builtin_prefetch((const void*)(Bp + (size_t)16 * DFF), 0, 0);
    acc00 = wmma_f32(a0, bv0, acc00);
    acc01 = wmma_f32(a0, bv1, acc01);
    acc10 = wmma_f32(a1, bv0, acc10);
    acc11 = wmma_f32(a1, bv1, acc11);
  }
  float bias0 = b1[(size_t)e * DFF + col0];
  float bias1 = b1[(size_t)e * DFF + col0 + 16];
#pragma unroll
  for (int i = 0; i < 8; ++i) {
    int r0 = rowBase + i + 8 * half;         // C/D layout: VGPR i, lanegroup adds 8
    int r1 = r0 + 16;
    Mid[(size_t)r0 * DFF + col0]      = gelu_exact(acc00[i] + bias0);
    Mid[(size_t)r0 * DFF + col0 + 16] = gelu_exact(acc01[i] + bias1);
    Mid[(size_t)r1 * DFF + col0]      = gelu_exact(acc10[i] + bias0);
    Mid[(size_t)r1 * DFF + col0 + 16] = gelu_exact(acc11[i] + bias1);
  }
}

// ---------------- 5) GEMM2 (Mid @ W2[e]) + bias -> per-row buffer ----------------
__global__ void moe_gemm2_kernel(const float* __restrict__ Mid,
                                 const float* __restrict__ W2,
                                 const float* __restrict__ b2,
                                 float* __restrict__ Buf2,
                                 const int* __restrict__ t2e,
                                 const int* __restrict__ meta) {
  int tile = blockIdx.x;
  if (tile >= meta[1]) return;
  int e = t2e[tile];
  if (e < 0) return;

  int lane = threadIdx.x & 31;
  int wave = threadIdx.x >> 5;
  int m = lane & 15, half = lane >> 4;
  int rowBase = tile * TILE_ROWS;
  int col0 = blockIdx.y * 256 + wave * 32 + m;

  const float* A0 = Mid + (size_t)(rowBase + m) * DFF + 2 * half;
  const float* A1 = A0 + (size_t)16 * DFF;
  const float* B = W2 + (size_t)e * DFF * H + (size_t)(2 * half) * H + col0;

  v8f acc00 = {0, 0, 0, 0, 0, 0, 0, 0};
  v8f acc01 = {0, 0, 0, 0, 0, 0, 0, 0};
  v8f acc10 = {0, 0, 0, 0, 0, 0, 0, 0};
  v8f acc11 = {0, 0, 0, 0, 0, 0, 0, 0};
#pragma unroll 4
  for (int k0 = 0; k0 < DFF; k0 += 4) {
    v2f a0 = *(const v2f*)(A0 + k0);
    v2f a1 = *(const v2f*)(A1 + k0);
    const float* Bp = B + (size_t)k0 * H;
    v2f bv0; bv0.x = Bp[0];  bv0.y = Bp[H];
    v2f bv1; bv1.x = Bp[16]; bv1.y = Bp[H + 16];
    __builtin_prefetch((const void*)(Bp + (size_t)16 * H), 0, 0);
    acc00 = wmma_f32(a0, bv0, acc00);
    acc01 = wmma_f32(a0, bv1, acc01);
    acc10 = wmma_f32(a1, bv0, acc10);
    acc11 = wmma_f32(a1, bv1, acc11);
  }
  float bias0 = b2[(size_t)e * H + col0];
  float bias1 = b2[(size_t)e * H + col0 + 16];
#pragma unroll
  for (int i = 0; i < 8; ++i) {
    int r0 = rowBase + i + 8 * half;
    int r1 = r0 + 16;
    Buf2[(size_t)r0 * H + col0]      = acc00[i] + bias0;
    Buf2[(size_t)r0 * H + col0 + 16] = acc01[i] + bias1;
    Buf2[(size_t)r1 * H + col0]      = acc10[i] + bias0;
    Buf2[(size_t)r1 * H + col0 + 16] = acc11[i] + bias1;
  }
}

// ---------------- 6) weighted combine (no atomics, deterministic) ----------------
__global__ void moe_combine_kernel(const float* __restrict__ Buf2,
                                   const float* __restrict__ wts,
                                   const int* __restrict__ rowof,
                                   float* __restrict__ out) {
  int gid = blockIdx.x * blockDim.x + threadIdx.x;  // T*H threads
  int t = gid / H, h = gid - t * H;
  int r0 = rowof[2 * t], r1 = rowof[2 * t + 1];
  out[gid] = wts[2 * t] * Buf2[(size_t)r0 * H + h] +
             wts[2 * t + 1] * Buf2[(size_t)r1 * H + h];
}

extern "C" void kernel_launch(void* const* d_in, const int* in_sizes, int n_in,
                              void* d_out, int out_size, void* d_ws, size_t ws_size,
                              hipStream_t stream) {
  const float* x  = (const float*)d_in[0];
  const float* Wg = (const float*)d_in[1];
  const float* bg = (const float*)d_in[2];
  const float* W1 = (const float*)d_in[3];
  const float* b1 = (const float*)d_in[4];
  const float* W2 = (const float*)d_in[5];
  const float* b2 = (const float*)d_in[6];
  float* out = (float*)d_out;

  char* ws = (char*)d_ws;
  int*   idx2  = (int*)(ws + OFF_IDX2);
  int*   rowof = (int*)(ws + OFF_ROWOF);
  int*   r2t   = (int*)(ws + OFF_R2T);
  int*   t2e   = (int*)(ws + OFF_T2E);
  int*   meta  = (int*)(ws + OFF_META);
  float* wts   = (float*)(ws + OFF_WTS);
  float* Xg    = (float*)(ws + OFF_XG);
  float* Mid   = (float*)(ws + OFF_MID);
  float* Buf2  = (float*)(ws + OFF_BUF2);

  moe_gate_kernel<<<T / 256, 256, 0, stream>>>(x, Wg, bg, idx2, wts);
  moe_route_kernel<<<1, 32, 0, stream>>>(idx2, r2t, rowof, t2e, meta);
  moe_gather_kernel<<<(RMAX * H) / 256, 256, 0, stream>>>(x, r2t, Xg);
  moe_gemm1_kernel<<<dim3(TILES_MAX, DFF / 256), 256, 0, stream>>>(Xg, W1, b1, Mid, t2e, meta);
  moe_gemm2_kernel<<<dim3(TILES_MAX, H / 256), 256, 0, stream>>>(Mid, W2, b2, Buf2, t2e, meta);
  moe_combine_kernel<<<(T * H) / 256, 256, 0, stream>>>(Buf2, wts, rowof, out);
}